// SparseLayer_11879879543150
// MI455X (gfx1250) — compile-verified
//
#include <hip/hip_runtime.h>
#include <stdint.h>

#define WAVES_PER_BLOCK 8
#define SCATTER_BLOCK (WAVES_PER_BLOCK * 32)

// ---------------------------------------------------------------------------
// Zero fill (float4 granularity)
// ---------------------------------------------------------------------------
__global__ void zero_f4_kernel(float4* __restrict__ p, long n4) {
    long i = (long)blockIdx.x * blockDim.x + threadIdx.x;
    long stride = (long)gridDim.x * blockDim.x;
    float4 z = make_float4(0.f, 0.f, 0.f, 0.f);
    for (; i < n4; i += stride) p[i] = z;
}

// ---------------------------------------------------------------------------
// Tiled transpose: src is R rows x C cols  ->  dst is C rows x R cols
// 32x32 tile, 32x8 threads, LDS padded to kill bank conflicts.
// ---------------------------------------------------------------------------
__global__ __launch_bounds__(256) void transpose_kernel(
    const float* __restrict__ src, float* __restrict__ dst, int R, int C) {
    __shared__ float tile[32][33];
    int c0 = blockIdx.x * 32;
    int r0 = blockIdx.y * 32;
#pragma unroll
    for (int k = 0; k < 4; ++k) {
        int r = r0 + threadIdx.y + k * 8;
        int c = c0 + threadIdx.x;
        if (r < R && c < C)
            tile[threadIdx.y + k * 8][threadIdx.x] = src[(long)r * C + c];
    }
    __syncthreads();
#pragma unroll
    for (int k = 0; k < 4; ++k) {
        int c = c0 + threadIdx.y + k * 8;
        int r = r0 + threadIdx.x;
        if (c < C && r < R)
            dst[(long)c * R + r] = tile[threadIdx.x][threadIdx.y + k * 8];
    }
}

// ---------------------------------------------------------------------------
// Async stage one 512B row (x_t[r][0:128]) into this wave's LDS slot.
// Each lane moves 16 bytes -> one global_load_async_to_lds_b128 per wave.
// Tracked by ASYNCcnt (CDNA5 async data-mover path).
// ---------------------------------------------------------------------------
typedef int v4i_t __attribute__((vector_size(4 * sizeof(int))));

__device__ __forceinline__ void async_stage_row(const float* gsrc, float* lbuf,
                                                int lane) {
    const float* g = gsrc + lane * 4;
    float* l = lbuf + lane * 4;
#if __has_builtin(__builtin_amdgcn_global_load_async_to_lds_b128)
    __builtin_amdgcn_global_load_async_to_lds_b128(
        (__attribute__((address_space(1))) v4i_t*)(g),
        (__attribute__((address_space(3))) v4i_t*)(l), 0, 0);
#else
    unsigned lds_off =
        (unsigned)(uintptr_t)(__attribute__((address_space(3))) void*)(l);
    asm volatile("global_load_async_to_lds_b128 %0, %1, off" ::"v"(lds_off),
                 "v"(g)
                 : "memory");
#endif
}

// ---------------------------------------------------------------------------
// Main scatter: one wave per nonzero per iteration.
//   stage x_t[rows[i+stride]] (async, next buffer) while processing i:
//   lane reads float4 from LDS, scales by w[i], 4x global_atomic_add_f32
//   into out_t[cols[i]][lane*4 .. lane*4+3] (contiguous 512B per nnz).
// out_t (51.2MB) and x_t (51.2MB) are L2-resident on MI455X (192MB L2).
// ---------------------------------------------------------------------------
__global__ __launch_bounds__(SCATTER_BLOCK) void spmm_scatter_kernel(
    const float* __restrict__ xt, const float* __restrict__ w,
    const int* __restrict__ rows, const int* __restrict__ cols,
    float* __restrict__ outt, int nnz) {
    __shared__ float lds[WAVES_PER_BLOCK * 2 * 128];
    const int lane = threadIdx.x & 31;
    const int wave = threadIdx.x >> 5;
    const long totalWaves = (long)gridDim.x * WAVES_PER_BLOCK;
    long i = (long)blockIdx.x * WAVES_PER_BLOCK + wave;
    if (i >= nnz) return;  // wave-uniform exit

    float* buf0 = &lds[(wave * 2 + 0) * 128];
    float* buf1 = &lds[(wave * 2 + 1) * 128];

    async_stage_row(xt + (long)rows[i] * 128, buf0, lane);
    int parity = 0;
    for (; i < nnz; i += totalWaves) {
        long nx = i + totalWaves;
        float* cur = parity ? buf1 : buf0;
        float* nxt = parity ? buf0 : buf1;
        if (nx < nnz) {
            async_stage_row(xt + (long)rows[nx] * 128, nxt, lane);
            asm volatile("s_wait_asynccnt 0x1" ::: "memory");
        } else {
            asm volatile("s_wait_asynccnt 0x0" ::: "memory");
        }
        float wv = w[i];
        long c = (long)cols[i];
        float4 v = *(const float4*)(cur + lane * 4);  // ds_load_b128
        float* dst = outt + c * 128 + lane * 4;
        atomicAdd(dst + 0, v.x * wv);  // global_atomic_add_f32 (no-return)
        atomicAdd(dst + 1, v.y * wv);
        atomicAdd(dst + 2, v.z * wv);
        atomicAdd(dst + 3, v.w * wv);
        parity ^= 1;
    }
}

// ---------------------------------------------------------------------------
// Fallback (workspace too small): direct strided scatter into d_out.
// ---------------------------------------------------------------------------
__global__ void spmm_direct_kernel(const float* __restrict__ x,
                                   const float* __restrict__ w,
                                   const int* __restrict__ rows,
                                   const int* __restrict__ cols,
                                   float* __restrict__ out, int nnz, int N,
                                   int M, int B) {
    const int lane = threadIdx.x & 31;
    const int wave = threadIdx.x >> 5;
    const long totalWaves = (long)gridDim.x * (blockDim.x >> 5);
    for (long i = (long)blockIdx.x * (blockDim.x >> 5) + wave; i < nnz;
         i += totalWaves) {
        int r = rows[i];
        long c = (long)cols[i];
        float wv = w[i];
        for (int b = lane; b < B; b += 32)
            atomicAdd(&out[(long)b * M + c], x[(long)b * N + r] * wv);
    }
}

// ---------------------------------------------------------------------------
extern "C" void kernel_launch(void* const* d_in, const int* in_sizes, int n_in,
                              void* d_out, int out_size, void* d_ws,
                              size_t ws_size, hipStream_t stream) {
    (void)n_in;
    const float* x = (const float*)d_in[0];
    const float* w = (const float*)d_in[1];
    const int* rows = (const int*)d_in[2];
    const int* cols = (const int*)d_in[3];
    float* out = (float*)d_out;

    const int B = 128;
    const int N = in_sizes[0] / B;
    const int NNZ = in_sizes[1];
    const int M = out_size / B;

    size_t need = ((size_t)N + (size_t)M) * (size_t)B * sizeof(float);
    dim3 tb(32, 8);

    if (d_ws != nullptr && ws_size >= need) {
        float* xt = (float*)d_ws;
        float* outt = xt + (size_t)N * B;

        // 1) zero out_t accumulator
        long n4 = ((long)M * B) / 4;
        long zb = (n4 + 255) / 256;
        int zgrid = (int)(zb < 4096 ? zb : 4096);
        if (zgrid < 1) zgrid = 1;
        zero_f4_kernel<<<zgrid, 256, 0, stream>>>((float4*)outt, n4);

        // 2) x [B,N] -> x_t [N,B]
        dim3 g1((N + 31) / 32, (B + 31) / 32);
        transpose_kernel<<<g1, tb, 0, stream>>>(x, xt, B, N);

        // 3) COO scatter with async-LDS double buffering
        long want = ((long)NNZ + WAVES_PER_BLOCK - 1) / WAVES_PER_BLOCK;
        int sblocks = (int)(want < 8192 ? want : 8192);
        if (sblocks < 1) sblocks = 1;
        spmm_scatter_kernel<<<sblocks, SCATTER_BLOCK, 0, stream>>>(
            xt, w, rows, cols, outt, NNZ);

        // 4) out_t [M,B] -> out [B,M]
        dim3 g2((B + 31) / 32, (M + 31) / 32);
        transpose_kernel<<<g2, tb, 0, stream>>>(outt, out, M, B);
    } else {
        long n4o = (long)out_size / 4;
        long zb = (n4o + 255) / 256;
        int zgrid = (int)(zb < 4096 ? zb : 4096);
        if (zgrid < 1) zgrid = 1;
        zero_f4_kernel<<<zgrid, 256, 0, stream>>>((float4*)out, n4o);
        long want = ((long)NNZ + 7) / 8;
        int sblocks = (int)(want < 8192 ? want : 8192);
        if (sblocks < 1) sblocks = 1;
        spmm_direct_kernel<<<sblocks, 256, 0, stream>>>(x, w, rows, cols, out,
                                                        NNZ, N, M, B);
    }
}